// MultiHeadAttention_76184129897179
// MI455X (gfx1250) — compile-verified
//
#include <hip/hip_runtime.h>

// Problem constants (from reference): B=4, T=4096, E=1024, HEAD=64, NH=16
#define B_N 4
#define T_N 4096
#define E_N 1024
#define H_N 64
#define NH_N 16
#define OUTW (H_N * NH_N)        /* 1024 */
#define ROWS_N (B_N * T_N)       /* 16384 total token rows */
#define LROW 18                  /* LDS row stride: even -> b64 reads, halves bank-disjoint */

typedef float v2f __attribute__((ext_vector_type(2)));
typedef float v8f __attribute__((ext_vector_type(8)));

static __device__ __forceinline__ v8f wmma_f32(v2f a, v2f b, v8f c) {
  // D(16x16,f32) = A(16x4,f32) * B(4x16,f32) + C  -- v_wmma_f32_16x16x4_f32
  return __builtin_amdgcn_wmma_f32_16x16x4_f32(false, a, false, b, (short)0, c,
                                               false, false);
}

// ---------------------------------------------------------------------------
// Kernel 0: one-time weight repack.  W[1024,64] -> Wt[64,1024] (x3 matrices)
// so that projection B-frags become one contiguous 8B load per lane.
// ---------------------------------------------------------------------------
__global__ __launch_bounds__(256) void transpose_w_kernel(
    const float* __restrict__ Wq, const float* __restrict__ Wk,
    const float* __restrict__ Wv, float* __restrict__ Wt) {
  const int idx = blockIdx.x * 256 + threadIdx.x;  // over 3*64*1024
  const int m = idx >> 16;                         // matrix 0..2
  const int c = (idx >> 10) & 63;                  // output column 0..63
  const int k = idx & 1023;                        // K index, fastest (coalesced writes)
  const float* W = (m == 0) ? Wq : (m == 1) ? Wk : Wv;
  Wt[(size_t)idx] = W[(size_t)k * H_N + c];
}

// ---------------------------------------------------------------------------
// Kernel 1: QKV projection.  X[16384,1024] @ W -> Q,K row-major [16384,64]
// and V transposed: Vt[64,16384] (so PV B-frags are contiguous 8B loads).
// One wave per 16-row tile; 12 accumulators; all frag loads are b64.
// ---------------------------------------------------------------------------
__global__ __launch_bounds__(256) void qkv_proj_kernel(
    const float* __restrict__ X, const float* __restrict__ Wt,
    float* __restrict__ Q, float* __restrict__ K, float* __restrict__ Vt) {
  const int lane = threadIdx.x & 31;
  const int wslot = threadIdx.x >> 5;
  const int tile = blockIdx.x * 8 + wslot;  // 0..1023, 16 rows each
  const int l15 = lane & 15;
  const int kOff = (lane >> 4) * 2;

  const v8f vzero = {0.f, 0.f, 0.f, 0.f, 0.f, 0.f, 0.f, 0.f};
  v8f acc[3][4];
#pragma unroll
  for (int m = 0; m < 3; ++m)
#pragma unroll
    for (int n = 0; n < 4; ++n) acc[m][n] = vzero;

  const float* xrow = X + (size_t)(tile * 16 + l15) * E_N;
  // Per-lane base into Wt for its B-frag column: Wt[m][n*16+l15][k]
  const float* wtb[3][4];
#pragma unroll
  for (int m = 0; m < 3; ++m)
#pragma unroll
    for (int n = 0; n < 4; ++n)
      wtb[m][n] = Wt + ((size_t)m * H_N + n * 16 + l15) * E_N;

  for (int k0 = 0; k0 < E_N; k0 += 4) {
    const int kb = k0 + kOff;
    const v2f a = *(const v2f*)(xrow + kb);  // 8B aligned (kb even)
#pragma unroll
    for (int m = 0; m < 3; ++m)
#pragma unroll
      for (int n = 0; n < 4; ++n) {
        const v2f bf = *(const v2f*)(wtb[m][n] + kb);  // contiguous 8B
        acc[m][n] = wmma_f32(a, bf, acc[m][n]);
      }
  }

  const int rbase = tile * 16 + (lane >> 4) * 8;
  // Q, K: row-major stores (coalesced)
#pragma unroll
  for (int n = 0; n < 4; ++n)
#pragma unroll
    for (int r = 0; r < 8; ++r) {
      Q[(size_t)(rbase + r) * H_N + n * 16 + l15] = acc[0][n][r];
      K[(size_t)(rbase + r) * H_N + n * 16 + l15] = acc[1][n][r];
    }
  // V: transposed store Vt[c][globalRow] (scattered, one-time, L2-resident)
#pragma unroll
  for (int n = 0; n < 4; ++n)
#pragma unroll
    for (int r = 0; r < 8; ++r)
      Vt[(size_t)(n * 16 + l15) * ROWS_N + rbase + r] = acc[2][n][r];
}

// ---------------------------------------------------------------------------
// Kernel 2: causal flash attention, one wave per 16-query tile.
// All B-frags are single b64 loads.  Softmax denominator is accumulated on
// the MATRIX pipe via a 5th "ones-column" accumulator (no per-tile row-sum
// butterfly); only the row-max reduction stays on VALU.  P staged through a
// per-wave LDS tile; next K/V tiles prefetched (global_prefetch_b8).
// ---------------------------------------------------------------------------
__global__ __launch_bounds__(256) void flash_attn_kernel(
    const float* __restrict__ Q, const float* __restrict__ K,
    const float* __restrict__ Vt, float* __restrict__ out) {
  __shared__ float lds[8][16 * LROW];  // one 16x16 P tile per wave

  const int lane = threadIdx.x & 31;
  const int wslot = threadIdx.x >> 5;
  const int gw = blockIdx.x * 8 + wslot;  // 0..1023
  const int b = gw >> 8;                  // batch 0..3
  const int qt = 255 - (gw & 255);        // heavy (long) tiles first
  const int l15 = lane & 15;
  const int half = lane >> 4;
  const int kOff = half * 2;

  const size_t base = (size_t)b * T_N * H_N;
  const size_t bt0 = (size_t)b * T_N;  // batch row offset for Vt columns
  const v8f vzero = {0.f, 0.f, 0.f, 0.f, 0.f, 0.f, 0.f, 0.f};

  // Preload Q A-frags for all 16 K-steps, folding in 1/sqrt(64) = 0.125.
  v2f qf[16];
  {
    const float* qrow = Q + base + (size_t)(qt * 16 + l15) * H_N;
#pragma unroll
    for (int j = 0; j < 16; ++j) {
      v2f t = *(const v2f*)(qrow + 4 * j + kOff);
      qf[j].x = t.x * 0.125f;
      qf[j].y = t.y * 0.125f;
    }
  }

  // Constant "ones" B-frag: column 0 of the 5th accumulator gathers row sums.
  v2f onesb;
  onesb.x = (l15 == 0) ? 1.0f : 0.0f;
  onesb.y = onesb.x;

  v8f o[4], o5;
#pragma unroll
  for (int n = 0; n < 4; ++n) o[n] = vzero;
  o5 = vzero;
  float mrow[8];
#pragma unroll
  for (int r = 0; r < 8; ++r) mrow[r] = -INFINITY;

  float* P = &lds[wslot][0];

  for (int st = 0; st <= qt; ++st) {
    // ---- prefetch next tile's K rows and Vt column lines ----
    if (st < qt) {
      __builtin_prefetch(K + base + (size_t)((st + 1) * 16 + l15) * H_N, 0, 1);
      __builtin_prefetch(Vt + (size_t)lane * ROWS_N + bt0 + (st + 1) * 16, 0, 1);
      __builtin_prefetch(Vt + (size_t)(lane + 32) * ROWS_N + bt0 + (st + 1) * 16, 0, 1);
    }

    // ---- S = Qs @ K^T for this 16x16 tile (16 b64 loads, 16 WMMAs) ----
    v8f s = vzero;
    const float* krow = K + base + (size_t)(st * 16 + l15) * H_N;
#pragma unroll
    for (int j = 0; j < 16; ++j) {
      const v2f kf = *(const v2f*)(krow + 4 * j + kOff);  // contiguous 8B
      s = wmma_f32(qf[j], kf, s);
    }

    // ---- causal mask (only the diagonal tile is partial) ----
    if (st == qt) {
#pragma unroll
      for (int r = 0; r < 8; ++r)
        if (l15 > r + 8 * half) s[r] = -INFINITY;
    }

    // ---- online softmax: row-max butterfly + exp (sum goes to matrix pipe) -
    float alpha[8];
#pragma unroll
    for (int r = 0; r < 8; ++r) {
      float rm = s[r];
      rm = fmaxf(rm, __shfl_xor(rm, 1, 32));
      rm = fmaxf(rm, __shfl_xor(rm, 2, 32));
      rm = fmaxf(rm, __shfl_xor(rm, 4, 32));
      rm = fmaxf(rm, __shfl_xor(rm, 8, 32));
      const float mn = fmaxf(mrow[r], rm);
      alpha[r] = __expf(mrow[r] - mn);
      mrow[r] = mn;
      s[r] = __expf(s[r] - mn);
    }

    // ---- rescale running output (and the denominator accumulator) ----
#pragma unroll
    for (int r = 0; r < 8; ++r) {
#pragma unroll
      for (int n = 0; n < 4; ++n) o[n][r] *= alpha[r];
      o5[r] *= alpha[r];
    }

    // ---- stage P: C/D layout -> LDS -> A-frag layout (in-order per wave) --
#pragma unroll
    for (int r = 0; r < 8; ++r) P[(r + 8 * half) * LROW + l15] = s[r];

    // ---- O += P @ V ; o5 += P @ e0 (row sums on the matrix pipe) ----
#pragma unroll
    for (int j = 0; j < 4; ++j) {
      const v2f pa = *(const v2f*)(P + l15 * LROW + 4 * j + kOff);  // ds b64
      const size_t key0 = bt0 + st * 16 + 4 * j + kOff;  // even -> 8B aligned
#pragma unroll
      for (int n = 0; n < 4; ++n) {
        const v2f vf = *(const v2f*)(Vt + (size_t)(n * 16 + l15) * ROWS_N + key0);
        o[n] = wmma_f32(pa, vf, o[n]);
      }
      o5 = wmma_f32(pa, onesb, o5);
    }
  }

  // ---- epilogue: extract denominators (column 0 of o5), normalize, write --
  float rl[8];
#pragma unroll
  for (int r = 0; r < 8; ++r) {
    const float lsum = __shfl(o5[r], half * 16, 32);  // lane 0 / lane 16
    rl[r] = 1.0f / lsum;
  }

  float* orow = out + ((size_t)b * T_N + (size_t)qt * 16) * OUTW;
#pragma unroll
  for (int n = 0; n < 4; ++n)
#pragma unroll
    for (int r = 0; r < 8; ++r) {
      const float val = o[n][r] * rl[r];
      float* dst = orow + (size_t)(r + 8 * half) * OUTW + n * 16 + l15;
#pragma unroll
      for (int h = 0; h < NH_N; ++h) dst[h * H_N] = val;
    }
}

// ---------------------------------------------------------------------------
extern "C" void kernel_launch(void* const* d_in, const int* in_sizes, int n_in,
                              void* d_out, int out_size, void* d_ws,
                              size_t ws_size, hipStream_t stream) {
  const float* X = (const float*)d_in[0];   // [4,4096,1024]
  const float* Wq = (const float*)d_in[1];  // [1024,64]
  const float* Wk = (const float*)d_in[2];
  const float* Wv = (const float*)d_in[3];
  // d_in[4] = num_heads (16), compile-time constant here.
  float* out = (float*)d_out;  // [4,4096,1024]

  const size_t qkv_elems = (size_t)ROWS_N * H_N;  // 1,048,576 each
  float* Qb = (float*)d_ws;
  float* Kb = Qb + qkv_elems;
  float* Vtb = Kb + qkv_elems;                    // transposed V [64,16384]
  float* Wtb = Vtb + qkv_elems;                   // transposed weights [3,64,1024]

  // Repack weights: 3*64*1024 elements / 256 threads = 768 blocks.
  transpose_w_kernel<<<768, 256, 0, stream>>>(Wq, Wk, Wv, Wtb);
  // 1024 waves each: 8 waves/block (256 threads) -> 128 blocks.
  qkv_proj_kernel<<<128, 256, 0, stream>>>(X, Wtb, Qb, Kb, Vtb);
  flash_attn_kernel<<<128, 256, 0, stream>>>(Qb, Kb, Vtb, out);
}